// VGAEncoder_33131377721458
// MI455X (gfx1250) — compile-verified
//
#include <hip/hip_runtime.h>

typedef __attribute__((ext_vector_type(2))) float v2f;
typedef __attribute__((ext_vector_type(8))) float v8f;

// ---------------------------------------------------------------------------
// Degree / normalization kernels
// ---------------------------------------------------------------------------
__global__ void deg_init_kernel(float* __restrict__ deg, int n) {
    int i = blockIdx.x * blockDim.x + threadIdx.x;
    if (i < n) deg[i] = 1.0f;                 // self-loop contributes 1 everywhere
}

__global__ void deg_accum_kernel(const int* __restrict__ dst,
                                 float* __restrict__ deg, int e) {
    int i = blockIdx.x * blockDim.x + threadIdx.x;
    if (i < e) atomicAdd(&deg[dst[i]], 1.0f);
}

__global__ void rsqrt_kernel(float* __restrict__ d, int n) {
    int i = blockIdx.x * blockDim.x + threadIdx.x;
    if (i < n) d[i] = rsqrtf(d[i]);           // deg >= 1 always (self-loops)
}

// ---------------------------------------------------------------------------
// fp32 WMMA GEMM: C[M x NC] = A[M x K] @ W[K x NC]
// One wave computes one 16x16 tile via V_WMMA_F32_16X16X4_F32 over K/4 steps.
// Requires M % 16 == 0, NC % 16 == 0, K % 4 == 0 (holds: 100000/128/96/64).
// ---------------------------------------------------------------------------
__global__ __launch_bounds__(128) void wmma_gemm_f32_kernel(
        const float* __restrict__ A, const float* __restrict__ W,
        float* __restrict__ C, int M, int K, int NC) {
    const int tilesN = NC >> 4;
    const int totalTiles = (M >> 4) * tilesN;
    const int wave = threadIdx.x >> 5;                 // 4 waves per block
    const int tile = blockIdx.x * 4 + wave;            // wave-uniform
    if (tile >= totalTiles) return;                    // uniform branch: EXEC stays full
    const int tm = tile / tilesN;
    const int tn = tile - tm * tilesN;
    const int lane = threadIdx.x & 31;
    const int half = lane >> 4;                        // 0: lanes 0-15, 1: lanes 16-31
    const int l16  = lane & 15;

    const float* arow = A + (size_t)(tm * 16 + l16) * K;   // A frag: M = l16
    const float* bcol = W + tn * 16 + l16;                 // B frag: N = l16

    v8f acc = {};
    for (int k0 = 0; k0 < K; k0 += 4) {
        // A 16x4 (f32): lanes 0-15 hold K = k0,k0+1 ; lanes 16-31 hold K = k0+2,k0+3
        v2f a, b;
        const int ka = k0 + 2 * half;
        a.x = arow[ka];
        a.y = arow[ka + 1];
        // B 4x16 (f32), mirrored layout: row K striped across lanes within a VGPR
        b.x = bcol[(size_t)ka * NC];
        b.y = bcol[(size_t)(ka + 1) * NC];
        acc = __builtin_amdgcn_wmma_f32_16x16x4_f32(
                  /*neg_a=*/false, a, /*neg_b=*/false, b,
                  /*c_mod=*/(short)0, acc, /*reuse_a=*/false, /*reuse_b=*/false);
    }

    // C/D layout: VGPR v -> row M = v + 8*half, col N = l16
    float* crow = C + (size_t)(tm * 16) * NC + tn * 16 + l16;
#pragma unroll
    for (int v = 0; v < 8; ++v)
        crow[(size_t)(v + 8 * half) * NC] = acc[v];
}

// ---------------------------------------------------------------------------
// out[i, :] = bias + dinv[i]^2 * h[i, :]    (self-loop message + bias, no atomics)
// ---------------------------------------------------------------------------
__global__ void init_bias_selfloop_kernel(
        const float* __restrict__ h, const float* __restrict__ dinv,
        const float* __restrict__ bias, float* __restrict__ out, int n, int ch) {
    const int chunks = ch >> 2;
    int idx = blockIdx.x * blockDim.x + threadIdx.x;
    int total = n * chunks;
    if (idx >= total) return;
    int i  = idx / chunks;
    int c4 = idx - i * chunks;
    float d = dinv[i];
    float s = d * d;
    float4 hv = reinterpret_cast<const float4*>(h + (size_t)i * ch)[c4];
    float4 bv = reinterpret_cast<const float4*>(bias)[c4];
    float4 o;
    o.x = bv.x + hv.x * s;
    o.y = bv.y + hv.y * s;
    o.z = bv.z + hv.z * s;
    o.w = bv.w + hv.w * s;
    reinterpret_cast<float4*>(out + (size_t)i * ch)[c4] = o;
}

// ---------------------------------------------------------------------------
// out[dst, :] += (dinv[src]*dinv[dst]) * h[src, :]   — L2-resident scatter-add
// ---------------------------------------------------------------------------
__global__ void edge_scatter_kernel(
        const float* __restrict__ h, const int* __restrict__ src,
        const int* __restrict__ dst, const float* __restrict__ dinv,
        float* __restrict__ out, int e, int ch) {
    const int chunks = ch >> 2;
    int idx = blockIdx.x * blockDim.x + threadIdx.x;
    int total = e * chunks;                       // <= 38.4M, fits int32
    if (idx >= total) return;
    int ei = idx / chunks;
    int c4 = idx - ei * chunks;
    int s = src[ei];
    int d = dst[ei];
    float nrm = dinv[s] * dinv[d];
    float4 hv = reinterpret_cast<const float4*>(h + (size_t)s * ch)[c4];
    float* o = out + (size_t)d * ch + (c4 << 2);
    atomicAdd(o + 0, hv.x * nrm);
    atomicAdd(o + 1, hv.y * nrm);
    atomicAdd(o + 2, hv.z * nrm);
    atomicAdd(o + 3, hv.w * nrm);
}

__global__ void relu_kernel(float* __restrict__ x, int n) {
    int i = blockIdx.x * blockDim.x + threadIdx.x;
    if (i < n) x[i] = fmaxf(x[i], 0.0f);
}

// ---------------------------------------------------------------------------
// Launch: GCN(128->96) + ReLU, then GCN(96->64) twice (mu, logvar)
// ---------------------------------------------------------------------------
extern "C" void kernel_launch(void* const* d_in, const int* in_sizes, int n_in,
                              void* d_out, int out_size, void* d_ws, size_t ws_size,
                              hipStream_t stream) {
    const float* x   = (const float*)d_in[0];
    const int*   ei  = (const int*)  d_in[1];   // [2, E]: row0 = src, row1 = dst
    const float* W1  = (const float*)d_in[2];
    const float* b1  = (const float*)d_in[3];
    const float* Wmu = (const float*)d_in[4];
    const float* bmu = (const float*)d_in[5];
    const float* Wlv = (const float*)d_in[6];
    const float* blv = (const float*)d_in[7];

    const int IN_CH = 128, MID_CH = 96, OUT_CH = 64;
    const int n = in_sizes[0] / IN_CH;          // 100000 (multiple of 16)
    const int e = in_sizes[1] / 2;              // 1600000
    const int* src = ei;
    const int* dst = ei + e;

    // Workspace layout (floats): dinv[n] | h[n*96] | agg[n*96]   (~77 MB)
    float* ws   = (float*)d_ws;
    float* dinv = ws;
    float* h    = dinv + n;                     // GEMM scratch (reused each layer)
    float* agg  = h + (size_t)n * MID_CH;       // post-aggregation hidden
    float* out_mu = (float*)d_out;
    float* out_lv = out_mu + (size_t)n * OUT_CH;

    const int T = 256;
    // --- symmetric normalization: dinv = rsqrt(deg + selfloop) ---
    deg_init_kernel <<<(n + T - 1) / T, T, 0, stream>>>(dinv, n);
    deg_accum_kernel<<<(e + T - 1) / T, T, 0, stream>>>(dst, dinv, e);
    rsqrt_kernel    <<<(n + T - 1) / T, T, 0, stream>>>(dinv, n);

    // --- layer 1: h = x @ W1 ; agg = scatter(h) + b1 ; relu ---
    {
        int tiles = (n / 16) * (MID_CH / 16);
        wmma_gemm_f32_kernel<<<(tiles + 3) / 4, 128, 0, stream>>>(x, W1, h, n, IN_CH, MID_CH);
        int tot = n * (MID_CH / 4);
        init_bias_selfloop_kernel<<<(tot + T - 1) / T, T, 0, stream>>>(h, dinv, b1, agg, n, MID_CH);
        int totE = e * (MID_CH / 4);
        edge_scatter_kernel<<<(totE + T - 1) / T, T, 0, stream>>>(h, src, dst, dinv, agg, e, MID_CH);
        relu_kernel<<<(n * MID_CH + T - 1) / T, T, 0, stream>>>(agg, n * MID_CH);
    }
    // --- mu head: h = agg @ Wmu ; out_mu = scatter(h) + bmu ---
    {
        int tiles = (n / 16) * (OUT_CH / 16);
        wmma_gemm_f32_kernel<<<(tiles + 3) / 4, 128, 0, stream>>>(agg, Wmu, h, n, MID_CH, OUT_CH);
        int tot = n * (OUT_CH / 4);
        init_bias_selfloop_kernel<<<(tot + T - 1) / T, T, 0, stream>>>(h, dinv, bmu, out_mu, n, OUT_CH);
        int totE = e * (OUT_CH / 4);
        edge_scatter_kernel<<<(totE + T - 1) / T, T, 0, stream>>>(h, src, dst, dinv, out_mu, e, OUT_CH);
    }
    // --- logvar head: h = agg @ Wlv ; out_lv = scatter(h) + blv ---
    {
        int tiles = (n / 16) * (OUT_CH / 16);
        wmma_gemm_f32_kernel<<<(tiles + 3) / 4, 128, 0, stream>>>(agg, Wlv, h, n, MID_CH, OUT_CH);
        int tot = n * (OUT_CH / 4);
        init_bias_selfloop_kernel<<<(tot + T - 1) / T, T, 0, stream>>>(h, dinv, blv, out_lv, n, OUT_CH);
        int totE = e * (OUT_CH / 4);
        edge_scatter_kernel<<<(totE + T - 1) / T, T, 0, stream>>>(h, src, dst, dinv, out_lv, e, OUT_CH);
    }
}